// Attention_76699525972201
// MI455X (gfx1250) — compile-verified
//
#include <hip/hip_runtime.h>
#include <math.h>

// Problem constants
namespace {
constexpr int kB     = 32;
constexpr int kL     = 4096;
constexpr int kH     = 512;
constexpr int kSplit = 16;                 // L-splits per batch -> 512 blocks in pass1
constexpr int kLc    = kL / kSplit;        // 256 rows per block
constexpr int kWaves = 8;                  // 256 threads = 8 wave32
constexpr int kRowsPerWave = kLc / kWaves; // 32
}

typedef __attribute__((ext_vector_type(2))) float v2f;
typedef __attribute__((ext_vector_type(8))) float v8f;

// ---------------------------------------------------------------------------
// Pass 1: stream context ONCE. Per (batch, L-split) block:
//   scores -> ws, plus online-softmax partials (max, sumexp, weighted mix).
// ---------------------------------------------------------------------------
__global__ __launch_bounds__(256) void attn_pass1(
    const float* __restrict__ outp,    // [B,H] query
    const float* __restrict__ ctx,     // [B,L,H]
    float* __restrict__ scores,        // [B,L]
    float* __restrict__ Mb,            // [B*kSplit]
    float* __restrict__ Sb,            // [B*kSplit]
    float* __restrict__ P)             // [B*kSplit, H] partial mixes
{
  __shared__ float wm[kWaves];
  __shared__ float wsum[kWaves];
  __shared__ float wacc[kWaves * kH];

  const int blk  = blockIdx.x;
  const int b    = blk / kSplit;
  const int s    = blk % kSplit;
  const int tid  = threadIdx.x;
  const int w    = tid >> 5;
  const int lane = tid & 31;

  // Preload query: lane holds h = ch*128 + lane*4 .. +3
  const float* q = outp + b * kH;
  float4 qv[4];
#pragma unroll
  for (int ch = 0; ch < 4; ++ch)
    qv[ch] = *(const float4*)(q + ch * 128 + lane * 4);

  float m = -INFINITY, ssum = 0.0f;
  float4 acc[4];
#pragma unroll
  for (int ch = 0; ch < 4; ++ch) { acc[ch].x = acc[ch].y = acc[ch].z = acc[ch].w = 0.0f; }

  const float* ctxB = ctx + (size_t)b * kL * kH;

  for (int i = 0; i < kRowsPerWave; ++i) {
    const int lrow = s * kLc + i * kWaves + w;
    const float* row = ctxB + (size_t)lrow * kH;

    float4 cv[4];
    float dot = 0.0f;
#pragma unroll
    for (int ch = 0; ch < 4; ++ch) {
      float4 c4 = *(const float4*)(row + ch * 128 + lane * 4);
      cv[ch] = c4;
      dot += c4.x * qv[ch].x + c4.y * qv[ch].y + c4.z * qv[ch].z + c4.w * qv[ch].w;
    }
    if (i + 1 < kRowsPerWave)
      __builtin_prefetch(row + kWaves * kH + lane * 16, 0, 0);  // global_prefetch_b8

    // wave32 butterfly reduction; all lanes end with the full dot
#pragma unroll
    for (int off = 16; off; off >>= 1)
      dot += __shfl_xor(dot, off, 32);

    const float score = (dot == 0.0f) ? -INFINITY : dot;  // exact-zero mask
    if (lane == 0) scores[b * kL + lrow] = score;

    // online softmax update
    const float mn = fmaxf(m, score);
    float scale, p;
    if (mn == -INFINITY) { scale = 1.0f; p = 0.0f; }
    else { scale = __expf(m - mn); p = __expf(score - mn); }
    ssum = ssum * scale + p;
#pragma unroll
    for (int ch = 0; ch < 4; ++ch) {
      acc[ch].x = acc[ch].x * scale + p * cv[ch].x;
      acc[ch].y = acc[ch].y * scale + p * cv[ch].y;
      acc[ch].z = acc[ch].z * scale + p * cv[ch].z;
      acc[ch].w = acc[ch].w * scale + p * cv[ch].w;
    }
    m = mn;
  }

  if (lane == 0) { wm[w] = m; wsum[w] = ssum; }
#pragma unroll
  for (int ch = 0; ch < 4; ++ch)
    *(float4*)&wacc[w * kH + ch * 128 + lane * 4] = acc[ch];
  __syncthreads();

  // combine the 8 waves of this block
  float M = -INFINITY;
#pragma unroll
  for (int j = 0; j < kWaves; ++j) M = fmaxf(M, wm[j]);
  float f[kWaves];
#pragma unroll
  for (int j = 0; j < kWaves; ++j)
    f[j] = (wm[j] == -INFINITY) ? 0.0f : __expf(wm[j] - M);

#pragma unroll
  for (int r = 0; r < kH / 256; ++r) {
    const int h = tid + r * 256;
    float a = 0.0f;
#pragma unroll
    for (int j = 0; j < kWaves; ++j) a += wacc[j * kH + h] * f[j];
    P[(size_t)(b * kSplit + s) * kH + h] = a;
  }
  if (tid == 0) {
    float S = 0.0f;
#pragma unroll
    for (int j = 0; j < kWaves; ++j) S += wsum[j] * f[j];
    Mb[b * kSplit + s] = M;
    Sb[b * kSplit + s] = S;
  }
}

// ---------------------------------------------------------------------------
// Pass 2: per batch, merge the kSplit partials; write attn to d_out and
// combined = [mix, output] to workspace.
// ---------------------------------------------------------------------------
__global__ __launch_bounds__(256) void attn_pass2(
    const float* __restrict__ outp,     // [B,H]
    const float* __restrict__ scores,   // [B,L]
    const float* __restrict__ Mb,
    const float* __restrict__ Sb,
    const float* __restrict__ P,        // [B*kSplit, H]
    float* __restrict__ combined,       // [B, 2H]
    float* __restrict__ attn_out)       // [B, L]  (d_out + B*H)
{
  const int b   = blockIdx.x;
  const int tid = threadIdx.x;

  float M = -INFINITY;
#pragma unroll
  for (int s = 0; s < kSplit; ++s) M = fmaxf(M, Mb[b * kSplit + s]);

  float f[kSplit];
  float S = 0.0f;
#pragma unroll
  for (int s = 0; s < kSplit; ++s) {
    const float mv = Mb[b * kSplit + s];
    f[s] = (mv == -INFINITY) ? 0.0f : __expf(mv - M);
    S += Sb[b * kSplit + s] * f[s];
  }
  const float inv = (S > 0.0f) ? (1.0f / S) : 0.0f;  // all-masked row -> 0 (NaN fill)

  // mix and combined
#pragma unroll
  for (int r = 0; r < kH / 256; ++r) {
    const int h = tid + r * 256;
    float a = 0.0f;
#pragma unroll
    for (int s = 0; s < kSplit; ++s)
      a += P[(size_t)(b * kSplit + s) * kH + h] * f[s];
    combined[(size_t)b * (2 * kH) + h]      = a * inv;
    combined[(size_t)b * (2 * kH) + kH + h] = outp[b * kH + h];
  }

  // attention weights
  for (int i = 0; i < kL / 256; ++i) {
    const int l = i * 256 + tid;
    const float sc = scores[b * kL + l];
    const float a  = (inv == 0.0f) ? 0.0f : __expf(sc - M) * inv;
    attn_out[(size_t)b * kL + l] = a;
  }
}

// ---------------------------------------------------------------------------
// Pass 3: out = tanh(combined[32,1024] @ W^T[1024,512] + bias) via
// V_WMMA_F32_16X16X4_F32.  One wave per 16x16 output tile.
// A layout (16x4 f32): lanes 0-15 -> K=(0,1), lanes 16-31 -> K=(2,3), M=lane&15.
// B layout (4x16 f32): mirrored, N=lane&15.
// D layout: VGPR r -> M=r (lanes 0-15) / M=r+8 (lanes 16-31), N=lane&15.
// ---------------------------------------------------------------------------
__global__ __launch_bounds__(32) void final_gemm_wmma(
    const float* __restrict__ combined,  // [32, 1024]
    const float* __restrict__ W,         // [512, 1024]  (out = combined @ W^T)
    const float* __restrict__ bias,      // [512]
    float* __restrict__ out)             // [32, 512]
{
  const int lane  = threadIdx.x;
  const int mtile = blockIdx.x & 1;        // 32/16 = 2
  const int ntile = blockIdx.x >> 1;       // 512/16 = 32
  const int mrow  = mtile * 16 + (lane & 15);
  const int ncol  = ntile * 16 + (lane & 15);
  const int koff  = (lane >> 4) << 1;      // 0 or 2

  const float* arow = combined + (size_t)mrow * (2 * kH) + koff;
  const float* brow = W + (size_t)ncol * (2 * kH) + koff;

  v8f c0 = {};
  v8f c1 = {};
  for (int k = 0; k < 2 * kH; k += 8) {
    v2f a0 = *(const v2f*)(arow + k);
    v2f b0 = *(const v2f*)(brow + k);
    c0 = __builtin_amdgcn_wmma_f32_16x16x4_f32(false, a0, false, b0,
                                               (short)0, c0, false, false);
    v2f a1 = *(const v2f*)(arow + k + 4);
    v2f b1 = *(const v2f*)(brow + k + 4);
    c1 = __builtin_amdgcn_wmma_f32_16x16x4_f32(false, a1, false, b1,
                                               (short)0, c1, false, false);
  }
  v8f c = c0 + c1;

  const float bv   = bias[ncol];
  const int   orow = mtile * 16 + (lane >> 4) * 8;
#pragma unroll
  for (int r = 0; r < 8; ++r)
    out[(size_t)(orow + r) * kH + ncol] = tanhf(c[r] + bv);
}

// ---------------------------------------------------------------------------
extern "C" void kernel_launch(void* const* d_in, const int* in_sizes, int n_in,
                              void* d_out, int out_size, void* d_ws, size_t ws_size,
                              hipStream_t stream) {
  const float* outp = (const float*)d_in[0];  // [B,H]
  const float* ctx  = (const float*)d_in[1];  // [B,L,H]
  const float* W    = (const float*)d_in[2];  // [H,2H]
  const float* bias = (const float*)d_in[3];  // [H]
  float* out = (float*)d_out;                 // [B*H] out ++ [B*L] attn

  // workspace layout (floats)
  float* ws       = (float*)d_ws;
  float* scores   = ws;                              // B*L      = 131072
  float* Mb       = scores + kB * kL;                // B*kSplit = 512
  float* Sb       = Mb + kB * kSplit;                // 512
  float* P        = Sb + kB * kSplit;                // B*kSplit*H = 262144
  float* combined = P + (size_t)kB * kSplit * kH;    // B*2H = 32768

  float* out_final = out;              // [32,512]
  float* attn_out  = out + kB * kH;    // [32,4096]

  attn_pass1<<<kB * kSplit, 256, 0, stream>>>(outp, ctx, scores, Mb, Sb, P);
  attn_pass2<<<kB, 256, 0, stream>>>(outp, scores, Mb, Sb, P, combined, attn_out);
  final_gemm_wmma<<<2 * (kH / 16), 32, 0, stream>>>(combined, W, bias, out_final);
}